// PresymplecticSoftmaxAttention_82626580840683
// MI455X (gfx1250) — compile-verified
//
#include <hip/hip_runtime.h>
#include <math.h>

typedef __attribute__((ext_vector_type(2))) float v2f;
typedef __attribute__((ext_vector_type(8))) float v8f;

namespace {

constexpr int Bc = 4;
constexpr int Tc = 2048;
constexpr int Cc = 768;
constexpr int NTOT = Bc * Tc * Cc;     // 6291456
constexpr int NROW = Bc * Tc;          // 8192
constexpr float SCALE_NH = 1.0f / 96.0f;   // (1/sqrt(64)) / 12
constexpr float TAU = 0.5f;
constexpr int KCH = 64;                // K / c chunk staged in LDS
constexpr int BP  = 68;                // lB pitch (floats), 16B-aligned rows
constexpr int ACP = 772;               // accumulator pitch
constexpr int SP  = 20;                // M-tile pitch

__device__ __forceinline__ float cube(float t) { return t * t * t; }

__device__ __forceinline__ v8f wmma_f32(v2f a, v2f b, v8f c) {
  // D = A(16x4) * B(4x16) + C, full fp32
  return __builtin_amdgcn_wmma_f32_16x16x4_f32(false, a, false, b, (short)0, c,
                                               false, false);
}

// ---- CDNA5 async global->LDS copy (ASYNCcnt-tracked, no VGPR round-trip) ----
__device__ __forceinline__ void async_copy_b128(void* lds, const void* gptr) {
  unsigned loff = (unsigned)(uintptr_t)lds;   // low 32b of generic = LDS offset
  asm volatile("global_load_async_to_lds_b128 %0, %1, off"
               :
               : "v"(loff), "v"(gptr)
               : "memory");
}
__device__ __forceinline__ void wait_async0() {
  asm volatile("s_wait_asynccnt 0x0" ::: "memory");
}

// stage one 16 x KCH chunk of Xb rows [s0..s0+15], cols [cc..cc+KCH) into buf
__device__ __forceinline__ void stage_chunk(float* buf, const float* Xb,
                                            int s0, int cc, int tid) {
#pragma unroll
  for (int i = tid * 4; i < 16 * KCH; i += 128 * 4) {
    int r = i >> 6, c = i & (KCH - 1);
    async_copy_b128(&buf[r * BP + c], &Xb[(size_t)(s0 + r) * Cc + cc + c]);
  }
}

// ---------------------------------------------------------------------------
// init: Pi = tk^3 * Pk ; bPi = Pi ; X = Xk ; bX = Xk
// ---------------------------------------------------------------------------
__global__ __launch_bounds__(256)
void k_init(const float* __restrict__ Xk, const float* __restrict__ Pk,
            float* __restrict__ X, float* __restrict__ Pi,
            float* __restrict__ bX, float* __restrict__ bPi,
            const int* __restrict__ kp) {
  unsigned i = blockIdx.x * 256u + threadIdx.x;
  float tk = 1.0f + (float)(*kp);
  float Lam = cube(tk);
  float pv = Lam * Pk[i];
  float xv = Xk[i];
  Pi[i] = pv; bPi[i] = pv;
  X[i] = xv;  bX[i] = xv;
}

// ---------------------------------------------------------------------------
// Fused LayerNorm(src)*w -> xn   and   asq[row] = sum_c pst^2
// ---------------------------------------------------------------------------
__global__ __launch_bounds__(256)
void k_ln_stats(const float* __restrict__ src, const float* __restrict__ pst,
                const float* __restrict__ w,
                float* __restrict__ xn, float* __restrict__ asq) {
  const int row = blockIdx.x;
  const int tid = threadIdx.x;
  const float* x = src + (size_t)row * Cc;
  const float* p = pst + (size_t)row * Cc;
  float* xo = xn + (size_t)row * Cc;

  float xv[3];
  float s1 = 0.f, s2 = 0.f, sp = 0.f;
#pragma unroll
  for (int i = 0; i < 3; ++i) {
    float v = x[tid + i * 256];
    xv[i] = v; s1 += v; s2 += v * v;
    float pv = p[tid + i * 256];
    sp += pv * pv;
  }
  __shared__ float r0[256], r1[256], r2[256];
  r0[tid] = s1; r1[tid] = s2; r2[tid] = sp;
  __syncthreads();
  for (int off = 128; off > 0; off >>= 1) {
    if (tid < off) {
      r0[tid] += r0[tid + off];
      r1[tid] += r1[tid + off];
      r2[tid] += r2[tid + off];
    }
    __syncthreads();
  }
  float m = r0[0] * (1.0f / Cc);
  float var = r1[0] * (1.0f / Cc) - m * m;
  float rstd = rsqrtf(var + 1e-5f);
#pragma unroll
  for (int i = 0; i < 3; ++i)
    xo[tid + i * 256] = (xv[i] - m) * rstd * w[tid + i * 256];
  if (tid == 0) asq[row] = r2[0];
}

// final LN with pre-scale lam = (tk+1)^-3, weight ln_v_w
__global__ __launch_bounds__(256)
void k_ln_final(const float* __restrict__ src, const float* __restrict__ w,
                float* __restrict__ out, const int* __restrict__ kp) {
  const int row = blockIdx.x;
  const int tid = threadIdx.x;
  float t1 = 2.0f + (float)(*kp);          // tk + 1
  float lam = 1.0f / cube(t1);
  const float* x = src + (size_t)row * Cc;
  float* xo = out + (size_t)row * Cc;

  float xv[3];
  float s1 = 0.f, s2 = 0.f;
#pragma unroll
  for (int i = 0; i < 3; ++i) {
    float v = lam * x[tid + i * 256];
    xv[i] = v; s1 += v; s2 += v * v;
  }
  __shared__ float r0[256], r1[256];
  r0[tid] = s1; r1[tid] = s2;
  __syncthreads();
  for (int off = 128; off > 0; off >>= 1) {
    if (tid < off) { r0[tid] += r0[tid + off]; r1[tid] += r1[tid + off]; }
    __syncthreads();
  }
  float m = r0[0] * (1.0f / Cc);
  float var = r1[0] * (1.0f / Cc) - m * m;
  float rstd = rsqrtf(var + 1e-5f);
#pragma unroll
  for (int i = 0; i < 3; ++i)
    xo[tid + i * 256] = (xv[i] - m) * rstd * w[tid + i * 256];
}

// ---------------------------------------------------------------------------
// z-pass: causal rowsum of E = exp(clamp(Xn Xn^T / 96)) via fp32 WMMA.
// Block = 128 thr (4 waves) handles 16 t-rows; waves split K of the gram.
// Double-buffered async global->LDS staging of the 16 x 64 B chunks.
// ---------------------------------------------------------------------------
__global__ __launch_bounds__(128)
void k_zpass(const float* __restrict__ xn, const float* __restrict__ asq,
             float* __restrict__ zv, float* __restrict__ sv,
             const int* __restrict__ kp, float t_off) {
  const int blk = blockIdx.x;
  const int b = blk / (Tc / 16);
  const int tb = blk % (Tc / 16);
  const int t0 = tb * 16;
  const float* Xb = xn + (size_t)b * Tc * Cc;

  __shared__ float lB[2][16 * BP];     // double-buffered B chunk
  __shared__ float lSp[3 * 256];       // wave partial S tiles
  __shared__ float zsh[16];

  const int tid = threadIdx.x;
  const int w = tid >> 5;
  const int lane = tid & 31;
  const int half = lane >> 4;
  const int l16 = lane & 15;

  float tk = 1.0f + (float)(*kp) + t_off;
  float lam = 1.0f / cube(tk);

  const float* At = Xb + (size_t)(t0 + l16) * Cc;   // this lane's A row

  float zacc[8];
#pragma unroll
  for (int j = 0; j < 8; ++j) zacc[j] = 0.f;

  const int nst = tb + 1;
  for (int st = 0; st < nst; ++st) {
    const int s0 = st * 16;
    stage_chunk(lB[0], Xb, s0, 0, tid);
    v8f Sacc = {};
    for (int cc = 0; cc < Cc; cc += KCH) {
      const int cur = (cc / KCH) & 1;
      wait_async0();
      __syncthreads();
      if (cc + KCH < Cc) stage_chunk(lB[cur ^ 1], Xb, s0, cc + KCH, tid);
      const float* bufc = lB[cur];
      const int kbase = w * (KCH / 4);               // this wave's K quarter
#pragma unroll
      for (int kk = 0; kk < KCH / 4; kk += 4) {
        int kl = kbase + kk;
        v2f a = *(const v2f*)&At[cc + kl + 2 * half];
        v2f bb = *(const v2f*)&bufc[l16 * BP + kl + 2 * half];
        Sacc = wmma_f32(a, bb, Sacc);
      }
    }
    __syncthreads();
    if (w > 0) {
#pragma unroll
      for (int j = 0; j < 8; ++j) lSp[(w - 1) * 256 + j * 32 + lane] = Sacc[j];
    }
    __syncthreads();
    if (w == 0) {
#pragma unroll
      for (int j = 0; j < 8; ++j) {
        float sfull = Sacc[j] + lSp[0 * 256 + j * 32 + lane] +
                      lSp[1 * 256 + j * 32 + lane] + lSp[2 * 256 + j * 32 + lane];
        int m = j + 8 * half;               // C/D layout: row
        int tg = t0 + m, sg = s0 + l16;     // col = l16
        if (sg <= tg) {
          float sc = fminf(fmaxf(sfull * SCALE_NH, -60.0f), 60.0f);
          zacc[j] += __expf(sc);
        }
      }
    }
  }

  if (w == 0) {
#pragma unroll
    for (int j = 0; j < 8; ++j) {
      float v = zacc[j];
      v += __shfl_xor(v, 1, 32);
      v += __shfl_xor(v, 2, 32);
      v += __shfl_xor(v, 4, 32);
      v += __shfl_xor(v, 8, 32);
      if (l16 == 0) zsh[j + 8 * half] = v;
    }
  }
  __syncthreads();
  if (tid < 16) {
    int row = b * Tc + t0 + tid;
    float z = fmaxf(zsh[tid] * (1.0f / Tc), 1e-8f);
    zv[row] = z;
    sv[row] = (lam * lam * asq[row]) / (z * z + 1e-8f);
  }
}

// ---------------------------------------------------------------------------
// core-pass: acc[16 x C] = sum_{s<=t} E*(s_t+s_s-2) @ Xn_s ; P += tau*Lam/(2T)*acc
// Gram as in z-pass; M built by wave0; second WMMA matmul splits c-tiles;
// all staging via double-buffered async global->LDS.
// ---------------------------------------------------------------------------
__global__ __launch_bounds__(128)
void k_core(const float* __restrict__ xn, const float* __restrict__ sv,
            float* __restrict__ pout, const int* __restrict__ kp, float t_off) {
  const int blk = blockIdx.x;
  const int b = blk / (Tc / 16);
  const int tb = blk % (Tc / 16);
  const int t0 = tb * 16;
  const float* Xb = xn + (size_t)b * Tc * Cc;

  __shared__ float lAcc[16 * ACP];     // 16 x 768 fp32 accumulator (~49KB)
  __shared__ float lB[2][16 * BP];     // double-buffered staged chunk
  __shared__ float lS[16 * SP];        // M tile, row-major for A-frags
  __shared__ float lSp[3 * 256];
  __shared__ float lsvt[16];

  const int tid = threadIdx.x;
  const int w = tid >> 5;
  const int lane = tid & 31;
  const int half = lane >> 4;
  const int l16 = lane & 15;

  float tk = 1.0f + (float)(*kp) + t_off;
  float Lam = cube(tk);
  float coef = TAU * Lam / (2.0f * Tc);

  const float* At = Xb + (size_t)(t0 + l16) * Cc;   // this lane's A row

  for (int i = tid * 4; i < 16 * Cc; i += 128 * 4) {
    int r = i / Cc, c = i % Cc;
    float4 z4 = make_float4(0.f, 0.f, 0.f, 0.f);
    *(float4*)&lAcc[r * ACP + c] = z4;
  }
  if (tid < 16) lsvt[tid] = sv[b * Tc + t0 + tid];
  __syncthreads();

  const int nst = tb + 1;
  for (int st = 0; st < nst; ++st) {
    const int s0 = st * 16;
    // ---- gram: S = Xn_t . Xn_s^T (waves split K)
    stage_chunk(lB[0], Xb, s0, 0, tid);
    v8f Sacc = {};
    for (int cc = 0; cc < Cc; cc += KCH) {
      const int cur = (cc / KCH) & 1;
      wait_async0();
      __syncthreads();
      if (cc + KCH < Cc) stage_chunk(lB[cur ^ 1], Xb, s0, cc + KCH, tid);
      const float* bufc = lB[cur];
      const int kbase = w * (KCH / 4);
#pragma unroll
      for (int kk = 0; kk < KCH / 4; kk += 4) {
        int kl = kbase + kk;
        v2f a = *(const v2f*)&At[cc + kl + 2 * half];
        v2f bb = *(const v2f*)&bufc[l16 * BP + kl + 2 * half];
        Sacc = wmma_f32(a, bb, Sacc);
      }
    }
    // prefetch first matmul2 chunk while we combine partials / build M
    stage_chunk(lB[0], Xb, s0, 0, tid);
    // ---- combine partials, build M tile (wave0)
    __syncthreads();
    if (w > 0) {
#pragma unroll
      for (int j = 0; j < 8; ++j) lSp[(w - 1) * 256 + j * 32 + lane] = Sacc[j];
    }
    __syncthreads();
    if (w == 0) {
      float svs = sv[b * Tc + s0 + l16];
#pragma unroll
      for (int j = 0; j < 8; ++j) {
        float sfull = Sacc[j] + lSp[0 * 256 + j * 32 + lane] +
                      lSp[1 * 256 + j * 32 + lane] + lSp[2 * 256 + j * 32 + lane];
        int m = j + 8 * half;
        int tg = t0 + m, sg = s0 + l16;
        float e = 0.f;
        if (sg <= tg) {
          float sc = fminf(fmaxf(sfull * SCALE_NH, -60.0f), 60.0f);
          e = __expf(sc);
        }
        lS[m * SP + l16] = e * (lsvt[m] + svs - 2.0f);
      }
    }
    // ---- second matmul: acc += M @ Xn_s ; waves split c-tiles (1 per wave)
    for (int cc = 0; cc < Cc; cc += KCH) {
      const int cur = (cc / KCH) & 1;
      wait_async0();
      __syncthreads();
      if (cc + KCH < Cc) stage_chunk(lB[cur ^ 1], Xb, s0, cc + KCH, tid);
      const float* bufc = lB[cur];
      int ctl = w * 16;                   // local col base within chunk
      int cg = cc + ctl;                  // global col base
      v8f acc;
#pragma unroll
      for (int j = 0; j < 8; ++j)
        acc[j] = lAcc[(j + 8 * half) * ACP + cg + l16];
#pragma unroll
      for (int kk = 0; kk < 16; kk += 4) {
        v2f a2 = *(const v2f*)&lS[l16 * SP + kk + 2 * half];
        v2f b2;
        b2.x = bufc[(kk + 2 * half) * BP + ctl + l16];
        b2.y = bufc[(kk + 1 + 2 * half) * BP + ctl + l16];
        acc = wmma_f32(a2, b2, acc);
      }
#pragma unroll
      for (int j = 0; j < 8; ++j)
        lAcc[(j + 8 * half) * ACP + cg + l16] = acc[j];
    }
    __syncthreads();
  }
  // ---- epilogue: P += coef * acc
  for (int i = tid; i < 16 * Cc; i += 128) {
    int r = i / Cc, c = i % Cc;
    size_t g = (size_t)(b * Tc + t0 + r) * Cc + c;
    pout[g] += coef * lAcc[r * ACP + c];
  }
}

// ---------------------------------------------------------------------------
// vel update: xout += tau * lam * P / z[row]
// ---------------------------------------------------------------------------
__global__ __launch_bounds__(256)
void k_vel(float* __restrict__ xout, const float* __restrict__ pst,
           const float* __restrict__ zv, const int* __restrict__ kp,
           float t_off) {
  unsigned i = blockIdx.x * 256u + threadIdx.x;
  float tk = 1.0f + (float)(*kp) + t_off;
  float lam = 1.0f / cube(tk);
  int row = i / Cc;
  xout[i] += TAU * lam * pst[i] / zv[row];
}

// ---------------------------------------------------------------------------
// rotation mixing, theta = 2
// ---------------------------------------------------------------------------
__global__ __launch_bounds__(256)
void k_rot(float* __restrict__ X, float* __restrict__ Pi,
           float* __restrict__ bX, float* __restrict__ bPi) {
  unsigned i = blockIdx.x * 256u + threadIdx.x;
  const float cn = -0.41614683654714241f;   // cos(2)
  const float sn = 0.90929742682568170f;    // sin(2)
  float x = X[i], p = Pi[i], bx = bX[i], bp = bPi[i];
  float dX = x - bx, dP = p - bp;
  float sX = x + bx, sP = p + bp;
  X[i]   = 0.5f * (sX + cn * dX + sn * dP);
  Pi[i]  = 0.5f * (sP - sn * dX + cn * dP);
  bX[i]  = 0.5f * (sX - cn * dX - sn * dP);
  bPi[i] = 0.5f * (sP + sn * dX - cn * dP);
}

} // namespace

// ---------------------------------------------------------------------------
extern "C" void kernel_launch(void* const* d_in, const int* in_sizes, int n_in,
                              void* d_out, int out_size, void* d_ws,
                              size_t ws_size, hipStream_t stream) {
  const float* Xk = (const float*)d_in[0];
  const float* Pk = (const float*)d_in[1];
  const float* ln_w = (const float*)d_in[2];
  const float* ln_v_w = (const float*)d_in[3];
  const int* kp = (const int*)d_in[4];

  float* X = (float*)d_out;            // output state X lives in d_out[0:N]
  float* out2 = X + NTOT;              // Pk1

  float* wPi = (float*)d_ws;
  float* wbX = wPi + NTOT;
  float* wbPi = wbX + NTOT;
  float* wXn = wbPi + NTOT;
  float* wasq = wXn + NTOT;
  float* wz = wasq + NROW;
  float* wsv = wz + NROW;

  const int EW = NTOT / 256;           // elementwise grid
  const int GT = Bc * (Tc / 16);       // tile grid (512)

  k_init<<<EW, 256, 0, stream>>>(Xk, Pk, X, wPi, wbX, wbPi, kp);

  auto phase = [&](const float* src, const float* pst, float toff, float* pout,
                   float* velx, bool dovel) {
    k_ln_stats<<<NROW, 256, 0, stream>>>(src, pst, ln_w, wXn, wasq);
    k_zpass<<<GT, 128, 0, stream>>>(wXn, wasq, wz, wsv, kp, toff);
    k_core<<<GT, 128, 0, stream>>>(wXn, wsv, pout, kp, toff);
    if (dovel)
      k_vel<<<EW, 256, 0, stream>>>(velx, pst, wz, kp, toff);
  };

  // phase 1: (t=tk, X, bPi) -> Pi += force ; bX += vel
  phase(X, wbPi, 0.0f, wPi, wbX, true);
  // phase 2: (t=tk+0.5, bX, Pi) -> bPi += force ; X += vel
  phase(wbX, wPi, 0.5f, wbPi, X, true);
  // rotation mixing
  k_rot<<<EW, 256, 0, stream>>>(X, wPi, wbX, wbPi);
  // phase 3: (t=tk+0.5, bX, Pi) -> bPi += force ; X += vel
  phase(wbX, wPi, 0.5f, wbPi, X, true);
  // phase 4: (t=tk+1, X, bPi) -> Pi += force  (bX vel not needed for output)
  phase(X, wbPi, 1.0f, wPi, nullptr, false);
  // Pk1 = LN((tk+1)^-3 * Pi, ln_v_w)
  k_ln_final<<<NROW, 256, 0, stream>>>(wPi, ln_v_w, out2, kp);
}